// SpectralAttention_10402410791715
// MI455X (gfx1250) — compile-verified
//
#include <hip/hip_runtime.h>
#include <hip/hip_bf16.h>

// ---------------------------------------------------------------------------
// SpectralAttention on MI455X (gfx1250, wave32, WMMA f32 16x16x4)
//
// Algebra: x_proj = x@pre_w.T + pre_b is only consumed via (@ filters), so we
// pre-fuse Wc = pre_w.T @ filters and bc = pre_b.filters, turning a 137-GFLOP
// GEMM + 268MB intermediate into a 4.3-GFLOP GEMM.
// Pipeline: fuse_weights -> WMMA GEMM (QV0) -> fused Z+chunked-scan (H) ->
//           WMMA GEMM multiply-into (H *= Qp) -> WMMA GEMM (out = H@o_w.T+o_b)
//
// GEMMs use double-buffered LDS tiles staged with GLOBAL_LOAD_ASYNC_TO_LDS
// (ASYNCcnt-tracked, partial-counter wait for copy/compute overlap) when the
// builtin is available, falling back to sync loads otherwise.
// ---------------------------------------------------------------------------

typedef __attribute__((ext_vector_type(2))) float v2f;
typedef __attribute__((ext_vector_type(8))) float v8f;

#define SEQ_LEN 2048
#define D_MODEL 1024
#define KF      32
#define BATCH   16
#define CP      528
#define BT      (BATCH * SEQ_LEN)   // 32768

#if defined(__gfx1250__) && __has_builtin(__builtin_amdgcn_global_load_async_to_lds_b128)
#define HAS_ASYNC_LDS 1
#else
#define HAS_ASYNC_LDS 0
#endif

#if HAS_ASYNC_LDS
// Builtin signature (from clang diagnostic): (v4i __device__* src,
// v4i __shared__* dst, imm offset, imm cpol). Flat pointers to __shared__
// carry the LDS byte offset in the low 32 bits; int->pointer casts build the
// address-space-qualified pointers without generic<->AS casts.
typedef int v4i __attribute__((vector_size(16)));
typedef __attribute__((address_space(1))) v4i* gp_v4i;   // global
typedef __attribute__((address_space(3))) v4i* lp_v4i;   // LDS
#define GPTR(p) ((gp_v4i)(unsigned long long)(p))
#define LPTR(p) ((lp_v4i)(unsigned int)(unsigned long long)(p))
#if __has_builtin(__builtin_amdgcn_s_wait_asynccnt)
#define ASYNC_WAIT(n) __builtin_amdgcn_s_wait_asynccnt(n)
#else
#define ASYNC_WAIT(n) asm volatile("s_wait_asynccnt %0" :: "n"(n) : "memory")
#endif
#else
#define ASYNC_WAIT(n) ((void)0)
#endif

// ---------------------------------------------------------------------------
// Kernel 1: WcT[n][d] = sum_i pre_w[i][d] * filt[i][n&31]   (n<32: q, n>=32: k)
//           bias_cat[n] = sum_i pre_b[i] * filt[i][n&31]
// ---------------------------------------------------------------------------
__global__ __launch_bounds__(256) void fuse_weights(
    const float* __restrict__ pre_w, const float* __restrict__ pre_b,
    const float* __restrict__ qf, const float* __restrict__ kf,
    float* __restrict__ WcT, float* __restrict__ bias_cat)
{
    const int n   = blockIdx.x;            // 0..63
    const int col = n & 31;
    const float* __restrict__ filt = (n < 32) ? qf : kf;
    const int tid = threadIdx.x;
    const int d0  = tid * 4;

    float acc0 = 0.f, acc1 = 0.f, acc2 = 0.f, acc3 = 0.f;
    for (int i = 0; i < SEQ_LEN; ++i) {
        const float f = filt[i * KF + col];                  // uniform -> SMEM
        const float4 w = *(const float4*)(pre_w + (size_t)i * D_MODEL + d0);
        acc0 += w.x * f; acc1 += w.y * f; acc2 += w.z * f; acc3 += w.w * f;
    }
    float* o = WcT + (size_t)n * D_MODEL + d0;
    o[0] = acc0; o[1] = acc1; o[2] = acc2; o[3] = acc3;

    __shared__ float red[256];
    float bacc = 0.f;
    for (int i = tid; i < SEQ_LEN; i += 256) bacc += pre_b[i] * filt[i * KF + col];
    red[tid] = bacc;
    __syncthreads();
    for (int s = 128; s > 0; s >>= 1) {
        if (tid < s) red[tid] += red[tid + s];
        __syncthreads();
    }
    if (tid == 0) bias_cat[n] = red[0];
}

// ---------------------------------------------------------------------------
// Tiled WMMA GEMM: C[M,N] = A[M,K] @ B[N,K]^T + bias[N]
//   MODE 0: C = result            MODE 1: C *= result  (multiply-into)
// Block 256 threads = 8 waves. Tile BM=64 x BN x BK=16, double-buffered LDS.
// 8 waves = 4 M-strips x 2 N-slabs; each wave owns BN/32 16x16 accumulators.
// Out-of-range B rows are loaded clamped (uniform ASYNCcnt), never stored.
// Requires: M % 64 == 0, K % 16 == 0. N guarded at store.
// ---------------------------------------------------------------------------
template <int MODE, int BN>
__global__ __launch_bounds__(256) void gemm_wmma(
    const float* __restrict__ A, const float* __restrict__ B,
    const float* __restrict__ bias, float* __restrict__ C,
    int M, int N, int Kd, int lda, int ldb, int ldc)
{
    constexpr int LS    = 18;            // padded LDS row stride (floats)
    constexpr int TPW   = BN / 32;       // 16x16 accumulators per wave (2 or 4)
    constexpr int BROWS = BN / 64;       // B tile rows staged per thread (1 or 2)
    constexpr int NIF   = 1 + BROWS;     // async instrs per thread per chunk

    __shared__ float As[2][64 * LS];
    __shared__ float Bs[2][BN * LS];

    const int tid  = threadIdx.x;
    const int wave = tid >> 5;
    const int lane = tid & 31;
    const int m0 = blockIdx.x * 64;
    const int n0 = blockIdx.y * BN;
    const int msub = (wave & 3) * 16;
    const int nsub = (wave >> 2) * (BN / 2);

    const int mlane = lane & 15;
    const int khalf = lane >> 4;         // 0 or 1

    // cooperative-load mapping: rows x 16 cols as float4/b128, one per thread
    const int lrow = tid >> 2;           // 0..63
    const int lc4  = (tid & 3) * 4;      // 0,4,8,12

    v8f acc[TPW] = {};

    // stage one BK=16 chunk into LDS buffer `buf`
    auto stage = [&](int k0, int buf) {
        const float* Ap = A + (size_t)(m0 + lrow) * lda + k0 + lc4;
#if HAS_ASYNC_LDS
        __builtin_amdgcn_global_load_async_to_lds_b128(
            GPTR(Ap), LPTR(&As[buf][lrow * LS + lc4]), 0, 0);
#else
        if (k0 + 16 < Kd) __builtin_prefetch(Ap + 16, 0, 1);
        const float4 av = *(const float4*)Ap;
        float* as = &As[buf][lrow * LS + lc4];
        as[0] = av.x; as[1] = av.y; as[2] = av.z; as[3] = av.w;
#endif
        #pragma unroll
        for (int r = 0; r < BROWS; ++r) {
            const int brow = lrow + r * 64;
            const int brow_g = (n0 + brow < N) ? (n0 + brow) : (N - 1); // clamp
            const float* Bp = B + (size_t)brow_g * ldb + k0 + lc4;
#if HAS_ASYNC_LDS
            __builtin_amdgcn_global_load_async_to_lds_b128(
                GPTR(Bp), LPTR(&Bs[buf][brow * LS + lc4]), 0, 0);
#else
            const float4 bv = *(const float4*)Bp;
            float* bs = &Bs[buf][brow * LS + lc4];
            bs[0] = bv.x; bs[1] = bv.y; bs[2] = bv.z; bs[3] = bv.w;
#endif
        }
    };

    const int nchunks = Kd >> 4;
    stage(0, 0);

    for (int c = 0; c < nchunks; ++c) {
        const int buf = c & 1;
        if (c + 1 < nchunks) {
            stage((c + 1) * 16, buf ^ 1);   // overlap next copy with this compute
            ASYNC_WAIT(NIF);                // chunk c landed; c+1 still in flight
        } else {
            ASYNC_WAIT(0);
        }
        __syncthreads();

        #pragma unroll
        for (int kk = 0; kk < 4; ++kk) {
            const int k = kk * 4;
            // A frag: lane m = lane&15 holds A[m][k + khalf*2 + {0,1}]
            const float* ap = &As[buf][(msub + mlane) * LS + k + khalf * 2];
            v2f a; a.x = ap[0]; a.y = ap[1];
            #pragma unroll
            for (int t = 0; t < TPW; ++t) {
                // B frag: vgpr v holds B[k + 2v + khalf][n], n = lane&15
                const float* bp = &Bs[buf][(nsub + t * 16 + mlane) * LS + k + khalf];
                v2f b; b.x = bp[0]; b.y = bp[2];
                acc[t] = __builtin_amdgcn_wmma_f32_16x16x4_f32(
                    false, a, false, b, (short)0, acc[t], false, false);
            }
        }
        __syncthreads();   // protect buf before it is re-staged next iteration
    }

    // ---- store: C layout vgpr r -> M = r + (lane>>4)*8, N = lane&15 ----
    const int rowbase = m0 + msub + khalf * 8;
    #pragma unroll
    for (int t = 0; t < TPW; ++t) {
        const int col = n0 + nsub + t * 16 + mlane;
        if (col < N) {
            const float bs = bias[col];
            #pragma unroll
            for (int r = 0; r < 8; ++r) {
                const size_t idx = (size_t)(rowbase + r) * ldc + col;
                const float v = acc[t][r] + bs;
                if (MODE == 0) C[idx] = v;
                else           C[idx] *= v;
            }
        }
    }
}

// ---------------------------------------------------------------------------
// Scan pass A: per (b, cp, time-chunk of 128): fused Z = V_i*V_j*scale*decay,
// local cumsum -> H, chunk total -> S[b][chunk][cp].
// ---------------------------------------------------------------------------
__global__ __launch_bounds__(256) void scan_local(
    const float* __restrict__ QV0, const float* __restrict__ decay,
    float* __restrict__ H, float* __restrict__ S)
{
    const int cp = blockIdx.x * 256 + threadIdx.x;
    if (cp >= CP) return;
    const int chunk = blockIdx.y;
    const int b = blockIdx.z;

    // decode upper-triangular (i,j) from cp
    int i = 0, base = 0;
    while (cp >= base + (KF - i)) { base += KF - i; ++i; }
    const int j = i + (cp - base);
    const float scale = (i == j) ? 1.0f : 1.41421356237309505f;

    float acc = 0.f;
    const int t0 = chunk * 128;
    for (int t = t0; t < t0 + 128; ++t) {
        const size_t bt = (size_t)b * SEQ_LEN + t;
        const float vi = QV0[bt * 64 + 32 + i];   // V lives in cols 32..63
        const float vj = QV0[bt * 64 + 32 + j];
        acc += vi * vj * scale * decay[t];
        H[bt * CP + cp] = acc;
    }
    S[((size_t)b * 16 + chunk) * CP + cp] = acc;
}

// ---------------------------------------------------------------------------
// Scan pass B: chunks 1..15 add the exclusive sum of prior chunk totals.
// ---------------------------------------------------------------------------
__global__ __launch_bounds__(256) void scan_fixup(
    float* __restrict__ H, const float* __restrict__ S)
{
    const int cp = blockIdx.x * 256 + threadIdx.x;
    if (cp >= CP) return;
    const int ch = blockIdx.y + 1;         // 1..15
    const int b = blockIdx.z;

    float off = 0.f;
    for (int c = 0; c < ch; ++c) off += S[((size_t)b * 16 + c) * CP + cp];

    const size_t btbase = (size_t)b * SEQ_LEN + ch * 128;
    for (int t = 0; t < 128; ++t)
        H[(btbase + t) * CP + cp] += off;
}

// ---------------------------------------------------------------------------
extern "C" void kernel_launch(void* const* d_in, const int* in_sizes, int n_in,
                              void* d_out, int out_size, void* d_ws, size_t ws_size,
                              hipStream_t stream) {
    const float* x     = (const float*)d_in[0];   // (16,2048,1024)
    const float* pre_w = (const float*)d_in[1];   // (2048,1024)
    const float* pre_b = (const float*)d_in[2];   // (2048,)
    const float* qf    = (const float*)d_in[3];   // (2048,32)
    const float* kf    = (const float*)d_in[4];   // (2048,32)
    const float* qp_w  = (const float*)d_in[5];   // (528,32)
    const float* qp_b  = (const float*)d_in[6];   // (528,)
    const float* o_w   = (const float*)d_in[7];   // (1024,528)
    const float* o_b   = (const float*)d_in[8];   // (1024,)
    const float* decay = (const float*)d_in[9];   // (2048,)
    float* out = (float*)d_out;                   // (16,2048,1024)

    char* ws = (char*)d_ws;
    float* WcT   = (float*)(ws);                          // 64*1024 f32 (256KB)
    float* biasc = (float*)(ws + (512u << 10));           // 64 f32
    float* QV0   = (float*)(ws + (1u << 20));             // 32768*64 f32 (8MB)
    float* S     = (float*)(ws + (10u << 20));            // 16*16*528 f32 (~540KB)
    float* H     = (float*)(ws + (12u << 20));            // 32768*528 f32 (69MB)

    // 1) fold pre-projection through the spectral filters
    fuse_weights<<<64, 256, 0, stream>>>(pre_w, pre_b, qf, kf, WcT, biasc);

    // 2) QV0 = x @ WcT^T + bias_cat   (M=32768, N=64, K=1024)
    gemm_wmma<0, 64><<<dim3(BT / 64, 1), 256, 0, stream>>>(
        x, WcT, biasc, QV0, BT, 64, D_MODEL, D_MODEL, D_MODEL, 64);

    // 3) fused outer-product + decay + chunked causal cumsum -> H
    scan_local<<<dim3(3, 16, BATCH), 256, 0, stream>>>(QV0, decay, H, S);
    scan_fixup<<<dim3(3, 15, BATCH), 256, 0, stream>>>(H, S);

    // 4) H *= (Q0 @ qp_w^T + qp_b)    (M=32768, N=528, K=32; A = QV0 cols 0..31)
    gemm_wmma<1, 128><<<dim3(BT / 64, (CP + 127) / 128), 256, 0, stream>>>(
        QV0, qp_w, qp_b, H, BT, CP, KF, 64, KF, CP);

    // 5) out = H @ o_w^T + o_b        (M=32768, N=1024, K=528)
    gemm_wmma<0, 128><<<dim3(BT / 64, D_MODEL / 128), 256, 0, stream>>>(
        H, o_w, o_b, out, BT, D_MODEL, CP, CP, CP, D_MODEL);
}